// Memory_17609365913744
// MI455X (gfx1250) — compile-verified
//
#include <hip/hip_runtime.h>

typedef __attribute__((ext_vector_type(2))) float v2f;
typedef __attribute__((ext_vector_type(8))) float v8f;
typedef __attribute__((ext_vector_type(4))) unsigned v4u;
typedef __attribute__((ext_vector_type(8))) int v8i;
typedef __attribute__((ext_vector_type(4))) int v4i;

#define EPSN 1e-8f

#if __has_builtin(__builtin_amdgcn_tensor_load_to_lds)
#define USE_TDM 1
#endif

// Padded LDS tile layout: 16 rows x (64 data + 4 pad) floats = 4352 B.
// Row stride 68 DWORDs => ds_load_b64 banks are 4*m mod 64: conflict-free.
#define ROWF 68
#define TILEF (16 * ROWF)
#define TPW 4          // tiles per wave
#define SIMT 128       // simKernel block = 4 waves

// ---------------------------------------------------------------------------
// Stage one 16x64 f32 tile (contiguous 4 KB of `memory`) into padded LDS.
// TDM path: one tensor_load_to_lds per tile, TENSORcnt tracked, HW padding.
// Fallback: 8x global_load_async_to_lds_b128 (512 B contiguous bursts).
// ---------------------------------------------------------------------------
__device__ __forceinline__ void stage_tile(const float* gsrc, float* ldst, int lane)
{
#ifdef USE_TDM
    unsigned ldsOff = (unsigned)(size_t)(void*)ldst;
    unsigned long long ga = (unsigned long long)(const void*)gsrc;
    v4u g0 = { 1u,                                   // count=1, user descriptor
               ldsOff,                               // lds_addr
               (unsigned)ga,                         // global_addr[31:0]
               ((unsigned)(ga >> 32) & 0x01FFFFFFu)  // global_addr[56:32]
                 | 0x80000000u };                    // type=2 ("image")
    // data_size=4B (2<<16) | pad_enable (1<<20) | pad_interval=5 -> 64 DWORDs
    // (5<<22) | pad_amount=3 -> 4 DWORDs (3<<25)
    v8i g1 = { 0x07520000,
               0x00400000,    // tensor_dim0 = 64 (bits 79:48 -> w1[31:16])
               0x00100000,    // tensor_dim1 = 16 (bits 111:80 -> w2[31:16])
               0x00400000,    // tile_dim0 = 64   (bits 127:112 -> w3[31:16])
               16,            // tile_dim1 = 16   (w4[15:0]), tile_dim2 = 0
               64,            // tensor_dim0_stride = 64
               0, 0 };
    v4i g2 = { 0, 0, 0, 0 };
    v4i g3 = { 0, 0, 0, 0 };
    v8i g4 = { 0, 0, 0, 0, 0, 0, 0, 0 };
    __builtin_amdgcn_tensor_load_to_lds(g0, g1, g2, g3, g4, 0);
    asm volatile("" ::: "memory");
#else
    // 8 issues x 32 lanes x 16B = 4 KB; each issue is a contiguous 512 B burst
    #pragma unroll
    for (int i = 0; i < 8; ++i) {
        int g   = i * 32 + lane;          // 16-byte chunk id within tile
        int row = g >> 4, ch = g & 15;
        unsigned ldsOff = (unsigned)(size_t)(void*)(ldst + row * ROWF + ch * 4);
        const float* ga = gsrc + row * 64 + ch * 4;
        asm volatile("global_load_async_to_lds_b128 %0, %1, off"
                     :: "v"(ldsOff), "v"(ga) : "memory");
    }
#endif
}

__device__ __forceinline__ void stage_wait(int pending)
{
#ifdef USE_TDM
#if __has_builtin(__builtin_amdgcn_s_wait_tensorcnt)
    if (pending) __builtin_amdgcn_s_wait_tensorcnt(1);
    else         __builtin_amdgcn_s_wait_tensorcnt(0);
#else
    if (pending) asm volatile("s_wait_tensorcnt 0x1" ::: "memory");
    else         asm volatile("s_wait_tensorcnt 0x0" ::: "memory");
#endif
#else
    if (pending) asm volatile("s_wait_asynccnt 0x8" ::: "memory");
    else         asm volatile("s_wait_asynccnt 0x0" ::: "memory");
#endif
    asm volatile("" ::: "memory");
}

// ---------------------------------------------------------------------------
// K1: fused GEMV (memory @ key) + row norms + exp(beta * cosine).
// One wave per 16-row tile (x TPW tiles, double-buffered TDM staging),
// V_WMMA_F32_16X16X4_F32 with the key chunk replicated across B columns.
// Row norm^2 rides on the same LDS reads (lane L: K%4 in {0,1};
// lane L+16: K%4 in {2,3}) -> combine with shfl_xor(16).
// ---------------------------------------------------------------------------
__global__ void simKernel(const float* __restrict__ mem,
                          const float* __restrict__ key,
                          const float* __restrict__ beta_p,
                          float* __restrict__ out0,
                          float* __restrict__ partials,
                          int n)
{
    __shared__ float red[SIMT];
    __shared__ float stage[(SIMT / 32) * 2 * TILEF];   // 4 waves x 2 bufs x 4352 B
    const int tid  = threadIdx.x;
    const int lane = tid & 31;
    const int wave = tid >> 5;

    // key norm (redundant per wave; trivially cheap, stays in cache)
    float k0 = key[lane], k1 = key[lane + 32];
    float ksq = k0 * k0 + k1 * k1;
    #pragma unroll
    for (int off = 16; off; off >>= 1) ksq += __shfl_xor(ksq, off, 32);
    const float keyNorm = fmaxf(sqrtf(ksq), EPSN);
    const float beta = beta_p[0];

    const int m    = lane & 15;           // row within tile
    const int kOff = (lane >> 4) * 2;     // 0 for lanes 0-15, 2 for 16-31
    float* wbuf = stage + wave * 2 * TILEF;
    const int tile0 = (blockIdx.x * (SIMT / 32) + wave) * TPW;

    stage_tile(mem + (size_t)tile0 * 16 * 64, wbuf, lane);

    float localE = 0.f;
    for (int kt = 0; kt < TPW; ++kt) {
        float* buf = wbuf + (kt & 1) * TILEF;
        if (kt + 1 < TPW)
            stage_tile(mem + (size_t)(tile0 + kt + 1) * 16 * 64,
                       wbuf + ((kt + 1) & 1) * TILEF, lane);
        stage_wait(kt + 1 < TPW);

        const int rowBase = (tile0 + kt) * 16;
        const float* rowPtr = buf + m * ROWF + kOff;
        v8f c = {};
        float sumsq = 0.f;
        #pragma unroll
        for (int t = 0; t < 16; ++t) {
            v2f a = *(const v2f*)(rowPtr + t * 4);   // ds_load_b64, bank-clean
            v2f b;
            b.x = key[t * 4 + kOff];
            b.y = key[t * 4 + kOff + 1];
            sumsq += a.x * a.x + a.y * a.y;
            c = __builtin_amdgcn_wmma_f32_16x16x4_f32(false, a, false, b,
                                                      (short)0, c, false, false);
        }
        sumsq += __shfl_xor(sumsq, 16, 32);          // full row |.|^2

        float carr[8];
        #pragma unroll
        for (int i = 0; i < 8; ++i) carr[i] = c[i];
        // D: lanes 0-15 hold M=0..7; lanes 16-31 hold M=8..15 (all cols equal).
        float xval = carr[(lane >= 24) ? (lane - 24) : (lane & 7)];
        float hi   = __shfl(xval, (lane & 7) + 24, 32);

        if (lane < 16) {
            float dot  = (lane < 8) ? carr[lane] : hi;
            float norm = fmaxf(sqrtf(sumsq), EPSN);
            float sim  = dot / (norm * keyNorm);
            float e    = __expf(sim * beta);   // |sim*beta| <= 2: no max-shift
            out0[rowBase + lane] = e;
            localE += e;
        }
    }

    red[tid] = localE;
    __syncthreads();
    #pragma unroll
    for (int off = SIMT / 2; off; off >>= 1) {
        if (tid < off) red[tid] += red[tid + off];
        __syncthreads();
    }
    if (tid == 0) partials[blockIdx.x] = red[0];
}

// deterministic fixed-order reduction of the block partials
__global__ void reduceSum(const float* __restrict__ partials,
                          float* __restrict__ sumE, int n)
{
    __shared__ float red[256];
    int tid = threadIdx.x;
    float s = 0.f;
    for (int k = tid; k < n; k += 256) s += partials[k];
    red[tid] = s;
    __syncthreads();
    #pragma unroll
    for (int off = 128; off; off >>= 1) {
        if (tid < off) red[tid] += red[tid + off];
        __syncthreads();
    }
    if (tid == 0) sumE[0] = red[0];
}

__global__ void normKernel(float* __restrict__ out0,
                           const float* __restrict__ sumE, int n)
{
    int i = blockIdx.x * 256 + threadIdx.x;
    if (i < n) out0[i] = out0[i] / sumE[0];
}

// ---------------------------------------------------------------------------
// K2: retention + usage + sortable key encode (streaming, float4 loads)
// ---------------------------------------------------------------------------
__global__ void retUsageKernel(const float* __restrict__ rw,
                               const float* __restrict__ fg,
                               const float* __restrict__ pu,
                               const float* __restrict__ wwt,
                               float* __restrict__ out1,
                               float* __restrict__ out2,
                               unsigned* __restrict__ keys,
                               unsigned* __restrict__ idx, int n)
{
    int i = blockIdx.x * 256 + threadIdx.x;
    if (i >= n) return;
    float4 r = ((const float4*)rw)[i];
    float ret = (1.f - r.x * fg[0]) * (1.f - r.y * fg[1]) *
                (1.f - r.z * fg[2]) * (1.f - r.w * fg[3]);
    float p = pu[i], w = wwt[i];
    float u = (p + w - p * w) * ret;
    out1[i] = ret;
    out2[i] = u;
    unsigned ub = __float_as_uint(u);
    ub = (ub & 0x80000000u) ? ~ub : (ub | 0x80000000u);  // monotone float->uint
    keys[i] = ub;
    idx[i]  = (unsigned)i;
}

// ---------------------------------------------------------------------------
// Stable LSD radix sort: 4 passes x 8 bits, 4096 elems / block.
// ---------------------------------------------------------------------------
__global__ void radixHist(const unsigned* __restrict__ keysIn,
                          unsigned* __restrict__ ghist, int shift, int nb)
{
    __shared__ unsigned hist[256];
    int tid = threadIdx.x, blk = blockIdx.x;
    hist[tid] = 0;
    __syncthreads();
    #pragma unroll
    for (int it = 0; it < 16; ++it) {
        unsigned key = keysIn[blk * 4096 + it * 256 + tid];
        atomicAdd(&hist[(key >> shift) & 255u], 1u);
    }
    __syncthreads();
    ghist[tid * nb + blk] = hist[tid];   // digit-major for stable global offsets
}

__global__ void radixScan(unsigned* __restrict__ ghist, int nChunks)
{
    __shared__ unsigned sdata[256];
    int tid = threadIdx.x;
    unsigned carry = 0;
    for (int c = 0; c < nChunks; ++c) {
        unsigned v = ghist[c * 256 + tid];
        sdata[tid] = v;
        __syncthreads();
        #pragma unroll
        for (int off = 1; off < 256; off <<= 1) {
            unsigned t = (tid >= off) ? sdata[tid - off] : 0u;
            __syncthreads();
            sdata[tid] += t;
            __syncthreads();
        }
        unsigned incl  = sdata[tid];
        unsigned total = sdata[255];
        ghist[c * 256 + tid] = carry + incl - v;   // exclusive
        carry += total;
        __syncthreads();
    }
}

__global__ void radixScatter(const unsigned* __restrict__ keysIn,
                             const unsigned* __restrict__ idxIn,
                             unsigned* __restrict__ keysOut,
                             unsigned* __restrict__ idxOut,
                             const unsigned* __restrict__ ghist,
                             int shift, int nb)
{
    __shared__ unsigned base[256];
    __shared__ unsigned cnt[256];
    __shared__ unsigned waveCnt[8 * 256];
    int tid = threadIdx.x, blk = blockIdx.x;
    base[tid] = ghist[tid * nb + blk];
    cnt[tid]  = 0;
    __syncthreads();
    const int lane = tid & 31, wave = tid >> 5;           // wave32
    const unsigned long long lowmask = ((1ull << lane) - 1ull);
    for (int it = 0; it < 16; ++it) {
        for (int k = tid; k < 8 * 256; k += 256) waveCnt[k] = 0;
        __syncthreads();
        int j = blk * 4096 + it * 256 + tid;
        unsigned key = keysIn[j], id = idxIn[j];
        unsigned d = (key >> shift) & 255u;
        // match mask of lanes with equal digit (8 ballots, wave32 low bits)
        unsigned long long m = 0xFFFFFFFFull;
        #pragma unroll
        for (int b = 0; b < 8; ++b) {
            unsigned long long bal = __ballot((d >> b) & 1u);
            m &= ((d >> b) & 1u) ? bal : ~bal;
        }
        unsigned laneRank = (unsigned)__popcll(m & lowmask);
        if (laneRank == 0)
            waveCnt[wave * 256 + d] = (unsigned)__popcll(m & 0xFFFFFFFFull);
        __syncthreads();
        unsigned preWave = 0;
        for (int w = 0; w < wave; ++w) preWave += waveCnt[w * 256 + d];
        unsigned dst = base[d] + cnt[d] + preWave + laneRank;
        keysOut[dst] = key;
        idxOut[dst]  = id;
        __syncthreads();
        unsigned add = 0;
        #pragma unroll
        for (int w = 0; w < 8; ++w) add += waveCnt[w * 256 + tid];
        cnt[tid] += add;
        __syncthreads();
    }
}

// ---------------------------------------------------------------------------
// Exclusive cumprod over sorted usage: per-block scan, block prefix, apply.
// ---------------------------------------------------------------------------
__global__ void blockCumprod(const unsigned* __restrict__ sortedIdx,
                             const float* __restrict__ usage,
                             float* __restrict__ tempProd,
                             float* __restrict__ blockTotal)
{
    __shared__ float sdata[256];
    int tid = threadIdx.x, blk = blockIdx.x;
    int base = blk * 4096 + tid * 16;
    float p[16];
    float run = 1.f;
    #pragma unroll
    for (int k = 0; k < 16; ++k) { run *= usage[sortedIdx[base + k]]; p[k] = run; }
    sdata[tid] = run;
    __syncthreads();
    #pragma unroll
    for (int off = 1; off < 256; off <<= 1) {
        float t = (tid >= off) ? sdata[tid - off] : 1.f;
        __syncthreads();
        sdata[tid] *= t;
        __syncthreads();
    }
    float excl = (tid == 0) ? 1.f : sdata[tid - 1];
    #pragma unroll
    for (int k = 0; k < 16; ++k) tempProd[base + k] = excl * p[k];
    if (tid == 0) blockTotal[blk] = sdata[255];
}

__global__ void blockPrefixKernel(const float* __restrict__ blockTotal,
                                  float* __restrict__ blockPrefix, int nb)
{
    if (threadIdx.x == 0 && blockIdx.x == 0) {
        float r = 1.f;
        for (int b = 0; b < nb; ++b) { blockPrefix[b] = r; r *= blockTotal[b]; }
    }
}

__global__ void allocKernel(const unsigned* __restrict__ sortedIdx,
                            const float* __restrict__ usage,
                            const float* __restrict__ tempProd,
                            const float* __restrict__ blockPrefix,
                            float* __restrict__ out3, int n)
{
    int j = blockIdx.x * 256 + threadIdx.x;
    if (j >= n) return;
    int b = j >> 12;
    float within = (j & 4095) ? tempProd[j - 1] : 1.f;
    float excl = blockPrefix[b] * within;                 // prod_{i<j} s[i]
    int pj = j ? j - 1 : 0;
    float sprev = usage[sortedIdx[pj]];                   // shifted = s[max(j-1,0)]
    out3[sortedIdx[j]] = (1.f - sprev) * excl;
}

// ---------------------------------------------------------------------------
extern "C" void kernel_launch(void* const* d_in, const int* in_sizes, int n_in,
                              void* d_out, int out_size, void* d_ws, size_t ws_size,
                              hipStream_t stream)
{
    const float* key   = (const float*)d_in[0];   // (64,)
    const float* mem   = (const float*)d_in[1];   // (N,64)
    const float* beta  = (const float*)d_in[2];   // (1,)
    const float* fg    = (const float*)d_in[3];   // (4,)
    const float* rw    = (const float*)d_in[4];   // (N,4)
    const float* pu    = (const float*)d_in[5];   // (N,)
    const float* wwt   = (const float*)d_in[6];   // (N,)

    const int N    = in_sizes[5];
    const int NB   = N / 4096;        // radix blocks (256 for N=2^20)
    const int SIMB = N / 256;         // simKernel blocks (4 waves x 4 tiles x 16)

    float* out  = (float*)d_out;
    float* out0 = out;                 // content_weighting
    float* out1 = out + (size_t)N;     // retention
    float* out2 = out + 2 * (size_t)N; // usage
    float* out3 = out + 3 * (size_t)N; // allocation

    unsigned* keysA = (unsigned*)d_ws;
    unsigned* idxA  = keysA + N;
    unsigned* keysB = idxA + N;
    unsigned* idxB  = keysB + N;
    float* tempProd = (float*)(idxB + N);
    unsigned* ghist = (unsigned*)(tempProd + N);
    float* partials = (float*)(ghist + 256 * (size_t)NB);
    float* blockTotal  = partials + SIMB;
    float* blockPrefix = blockTotal + NB;
    float* sumE        = blockPrefix + NB;

    // content lookup (TDM-staged WMMA GEMV + norms + exp), softmax, normalize
    simKernel<<<SIMB, SIMT, 0, stream>>>(mem, key, beta, out0, partials, N);
    reduceSum<<<1, 256, 0, stream>>>(partials, sumE, SIMB);
    normKernel<<<N / 256, 256, 0, stream>>>(out0, sumE, N);

    // retention + usage + sort keys
    retUsageKernel<<<N / 256, 256, 0, stream>>>(rw, fg, pu, wwt, out1, out2,
                                                keysA, idxA, N);

    // 4-pass stable LSD radix sort (key,index) ascending
    unsigned *kIn = keysA, *iIn = idxA, *kOut = keysB, *iOut = idxB;
    for (int p = 0; p < 4; ++p) {
        radixHist<<<NB, 256, 0, stream>>>(kIn, ghist, p * 8, NB);
        radixScan<<<1, 256, 0, stream>>>(ghist, NB);
        radixScatter<<<NB, 256, 0, stream>>>(kIn, iIn, kOut, iOut, ghist, p * 8, NB);
        unsigned* t;
        t = kIn; kIn = kOut; kOut = t;
        t = iIn; iIn = iOut; iOut = t;
    }
    // after 4 swaps sorted data is back in keysA/idxA (== kIn/iIn)

    // allocation weighting: two-level cumprod over sorted usage + scatter
    blockCumprod<<<NB, 256, 0, stream>>>(iIn, out2, tempProd, blockTotal);
    blockPrefixKernel<<<1, 64, 0, stream>>>(blockTotal, blockPrefix, NB);
    allocKernel<<<N / 256, 256, 0, stream>>>(iIn, out2, tempProd, blockPrefix,
                                             out3, N);
}